// SkeletonConvolution_71949292142897
// MI455X (gfx1250) — compile-verified
//
#include <hip/hip_runtime.h>
#include <stdint.h>

typedef __attribute__((ext_vector_type(2))) float v2f;
typedef __attribute__((ext_vector_type(8))) float v8f;

#define JOINTS   24
#define T        2048
#define CIN      384
#define COUT     384
#define NBLK     70      // nonzero 16x16 joint blocks in the skeleton adjacency
#define TN       32      // time-tile columns per workgroup
#define LDS_COLS 34      // TN + 2 halo
#define S2       48      // row stride in float2 units: 2*S2 % 64 == 32 -> conflict-free b64
#define NPAIR    (CIN/2) // 192 channel-pair rows
#define LDS_BYTES (NPAIR * S2 * 8)   // 73,728 B (dynamic shared)
#define FRAG_PER_BLK 768 // 3 taps * 4 k-chunks * 32 lanes * 2 vgprs

// CDNA5 async global->LDS copy path (ASYNCcnt) for the x tile fill.
#ifndef USE_ASYNC_LDS
#define USE_ASYNC_LDS 1
#endif

__constant__ int NEIGH_START[25] = {0,4,7,10,13,16,19,22,25,28,33,35,37,40,43,
                                    46,48,51,54,57,60,63,66,68,70};
__constant__ int NEIGH_FLAT[70] = {
  0,1,2,3,  1,0,4,  2,0,5,  3,0,6,  4,1,7,  5,2,8,  6,3,9,  7,4,10,  8,5,11,
  9,6,12,13,14,  10,7,  11,8,  12,9,15,  13,9,16,  14,9,17,  15,12,
  16,13,18,  17,14,19,  18,16,20,  19,17,21,  20,18,22,  21,19,23,  22,20,  23,21 };

// ---------------------------------------------------------------------------
// Kernel 0: gather masked weights into WMMA A-fragment layout.
// wp[bi][k][c][lane][v] = W[co = j*16 + lane%16][ci = ij*16 + 4c + 2*(lane/16) + v][k]
// ---------------------------------------------------------------------------
__global__ void pack_weights(const float* __restrict__ w, float* __restrict__ wp) {
  int e = blockIdx.x * blockDim.x + threadIdx.x;
  if (e >= NBLK * FRAG_PER_BLK) return;
  int bi   = e / FRAG_PER_BLK;
  int rem  = e - bi * FRAG_PER_BLK;
  int k    = rem >> 8;
  int c    = (rem >> 6) & 3;
  int lane = (rem >> 1) & 31;
  int v    = rem & 1;
  int j = 0;
  while (NEIGH_START[j + 1] <= bi) j++;   // out-joint owning this block
  int ij = NEIGH_FLAT[bi];                // in-joint (neighbor)
  int m  = lane & 15;
  int hi = lane >> 4;
  int co = j * 16 + m;
  int ci = ij * 16 + 4 * c + 2 * hi + v;
  wp[e] = w[(co * CIN + ci) * 3 + k];
}

// ---------------------------------------------------------------------------
// Kernel 1: block-sparse conv as chained V_WMMA_F32_16X16X4_F32.
// LDS layout: channel-pair interleaved, lds2[p][col] = float2{x[2p][t], x[2p+1][t]}
// so every B fragment is one ds_load_b64, bank-conflict free.
// ---------------------------------------------------------------------------
__global__ void __launch_bounds__(128)
skel_conv(const float* __restrict__ x, const float* __restrict__ wp,
          const float* __restrict__ bias, float* __restrict__ out) {
  extern __shared__ float lds[];          // LDS_BYTES, pair-interleaved x tile

  const int tid  = (int)threadIdx.x;
  const int tile = (int)blockIdx.x;
  const int b    = tile >> 6;             // 64 tiles per batch (T/TN)
  const int t0   = (tile & 63) * TN;

  // ---- cooperative fill of x tile (cols t0-1 .. t0+32), zero halo ----
  const float* xb = x + (size_t)b * CIN * T;
  for (int i = tid; i < CIN * LDS_COLS; i += 128) {
    int row = i / LDS_COLS;               // ci
    int col = i - row * LDS_COLS;
    int g   = t0 - 1 + col;
    float* dst = lds + ((((row >> 1) * S2 + col) << 1) + (row & 1));
    if (g >= 0 && g < T) {
#if USE_ASYNC_LDS
      unsigned ldsAddr = (unsigned)(uintptr_t)dst;
      const float* gp  = xb + (size_t)row * T + g;
      asm volatile("global_load_async_to_lds_b32 %0, %1, off"
                   :: "v"(ldsAddr), "v"(gp) : "memory");
#else
      *dst = xb[(size_t)row * T + g];
#endif
    } else {
      *dst = 0.0f;                        // zero temporal padding
    }
  }
#if USE_ASYNC_LDS
  asm volatile("s_wait_asynccnt 0x0" ::: "memory");
#endif
  __syncthreads();

  // ---- per-wave tile assignment (forced wave-uniform -> scalar control flow) ----
  const int lane  = tid & 31;
  const int n     = lane & 15;            // WMMA N / column within 16-wide tile
  const int hi    = lane >> 4;            // half-wave select
  const int wid_s = __builtin_amdgcn_readfirstlane(tid >> 5);
  const int ct    = wid_s & 1;            // which 16-column sub-tile
  const int jbase = (wid_s >> 1) * 12;    // joints [0,12) or [12,24)

  const v2f* ldsp    = (const v2f*)lds;
  const int  laneOff = hi * S2 + ct * 16 + n;   // per-lane part of the B address

  for (int jj = 0; jj < 12; ++jj) {
    const int j = jbase + jj;             // scalar
    v8f a0 = {0.f,0.f,0.f,0.f,0.f,0.f,0.f,0.f};
    v8f a1 = {0.f,0.f,0.f,0.f,0.f,0.f,0.f,0.f};

    const int s = NEIGH_START[j], e = NEIGH_START[j + 1];   // scalar bounds
    for (int ni = s; ni < e; ++ni) {
      const int ij = NEIGH_FLAT[ni];      // scalar
      const v2f* bp = ldsp + (size_t)(ij * 8) * S2 + laneOff;
      const float* wpb = wp + (size_t)ni * FRAG_PER_BLK + lane * 2;
#pragma unroll
      for (int k = 0; k < 3; ++k) {
        // A fragments: pre-packed, fully coalesced b64 loads (L2-resident)
        v2f A0 = *(const v2f*)(wpb + (k * 4 + 0) * 64);
        v2f A1 = *(const v2f*)(wpb + (k * 4 + 1) * 64);
        v2f A2 = *(const v2f*)(wpb + (k * 4 + 2) * 64);
        v2f A3 = *(const v2f*)(wpb + (k * 4 + 3) * 64);
        // B fragments: single ds_load_b64 each, constant LDS offsets
        v2f B0 = bp[0 * S2 + k];
        v2f B1 = bp[2 * S2 + k];
        v2f B2 = bp[4 * S2 + k];
        v2f B3 = bp[6 * S2 + k];
        // two independent accumulation chains
        a0 = __builtin_amdgcn_wmma_f32_16x16x4_f32(false, A0, false, B0, (short)0, a0, false, false);
        a1 = __builtin_amdgcn_wmma_f32_16x16x4_f32(false, A2, false, B2, (short)0, a1, false, false);
        a0 = __builtin_amdgcn_wmma_f32_16x16x4_f32(false, A1, false, B1, (short)0, a0, false, false);
        a1 = __builtin_amdgcn_wmma_f32_16x16x4_f32(false, A3, false, B3, (short)0, a1, false, false);
      }
    }

    // ---- bias + store (C/D layout: VGPR r -> M = r + 8*hi, lane -> N) ----
    v8f acc = a0 + a1;
    float* ob = out + ((size_t)b * COUT + j * 16) * T + t0 + ct * 16 + n;
#pragma unroll
    for (int r = 0; r < 8; ++r) {
      const int m = r + 8 * hi;
      __builtin_nontemporal_store(acc[r] + bias[j * 16 + m], ob + (size_t)m * T);
    }
  }
}

// ---------------------------------------------------------------------------
extern "C" void kernel_launch(void* const* d_in, const int* in_sizes, int n_in,
                              void* d_out, int out_size, void* d_ws, size_t ws_size,
                              hipStream_t stream) {
  const float* x    = (const float*)d_in[0];   // [64, 384, 2048]
  const float* w    = (const float*)d_in[1];   // [384, 384, 3]
  const float* bias = (const float*)d_in[2];   // [384]
  float* out = (float*)d_out;
  float* wp  = (float*)d_ws;                   // 70*768 floats = 215,040 B scratch

  const int packN = NBLK * FRAG_PER_BLK;
  pack_weights<<<(packN + 255) / 256, 256, 0, stream>>>(w, wp);

  const int tiles = 64 * (T / TN);             // 4096 workgroups
  skel_conv<<<tiles, 128, LDS_BYTES, stream>>>(x, wp, bias, out);
}